// QuanConv2D_65283502899387
// MI455X (gfx1250) — compile-verified
//
#include <hip/hip_runtime.h>
#include <math.h>

#define K     3
#define NQ    9
#define H     384
#define HOUT  128
#define WOUT  128
#define ROWF  (K * H)          // 1152 floats (4608 B) staged per block
#define PI_F  3.14159265358979323846f

// Closed form of the reference:
//   z_q(x)   = A_q*cos(pi*x) + B_q*sin(pi*x)
//   A_q      = cos(a0)cos(a2) - sin(a0)sin(a2)cos(a1)
//   B_q      = sin(a1)sin(a2),   a_i = weight[3q+i]*sqrt(2/5)
//   out[0]   = z_1*...*z_8
//   out[q>0] = z_0*...*z_q       (CNOT-ring parity sets)
__global__ __launch_bounds__(128) void quanconv_kernel(
    const float* __restrict__ x, const float* __restrict__ weight,
    float* __restrict__ out)
{
    __shared__ float rows[ROWF];     // 3 contiguous input rows
    __shared__ float sA[NQ], sB[NQ];

    const int t   = threadIdx.x;
    const int blk = blockIdx.x;      // b * HOUT + ho
    const int b   = blk >> 7;
    const int ho  = blk & 127;

    // ---- Stage 3 input rows (one contiguous 4608 B span) into LDS with
    //      CDNA5 async-to-LDS copies (ASYNCcnt-tracked, no VGPR staging).
    const float* gbase = x + ((size_t)b * H * H + (size_t)ho * K * H);
    {
        unsigned            l0 = (unsigned)(unsigned long long)(const void*)&rows[t * 4];
        unsigned long long  g0 = (unsigned long long)(const void*)(gbase + t * 4);
        asm volatile("global_load_async_to_lds_b128 %0, %1, off"
                     :: "v"(l0), "v"(g0) : "memory");
        unsigned            l1 = (unsigned)(unsigned long long)(const void*)&rows[(t + 128) * 4];
        unsigned long long  g1 = (unsigned long long)(const void*)(gbase + (t + 128) * 4);
        asm volatile("global_load_async_to_lds_b128 %0, %1, off"
                     :: "v"(l1), "v"(g1) : "memory");
        if (t < 32) {   // wave 0 fully active: chunks 256..287
            unsigned            l2 = (unsigned)(unsigned long long)(const void*)&rows[(t + 256) * 4];
            unsigned long long  g2 = (unsigned long long)(const void*)(gbase + (t + 256) * 4);
            asm volatile("global_load_async_to_lds_b128 %0, %1, off"
                         :: "v"(l2), "v"(g2) : "memory");
        }
    }

    // ---- Per-qubit constants, overlapped with the async copy.
    if (t < NQ) {
        const float WM = 0.63245553203367586640f;   // sqrt(2/5)
        float a0 = weight[t * 3 + 0] * WM;
        float a1 = weight[t * 3 + 1] * WM;
        float a2 = weight[t * 3 + 2] * WM;
        float s0, c0, s1, c1, s2, c2;
        __sincosf(a0, &s0, &c0);
        __sincosf(a1, &s1, &c1);
        __sincosf(a2, &s2, &c2);
        sA[t] = c0 * c2 - s0 * s2 * c1;
        sB[t] = s1 * s2;
    }

    asm volatile("s_wait_asynccnt 0" ::: "memory");
    __syncthreads();

    // ---- Per-thread: one output column (wo = t), 9 qubit z-values.
    const int wo = t;
    float z[NQ];
    #pragma unroll
    for (int i = 0; i < K; ++i) {
        #pragma unroll
        for (int j = 0; j < K; ++j) {
            const int q  = i * K + j;
            const float th = PI_F * rows[i * H + wo * K + j];
            float sn, cs;
            __sincosf(th, &sn, &cs);
            z[q] = sA[q] * cs + sB[q] * sn;
        }
    }

    float o[NQ];
    float pre = z[0];
    #pragma unroll
    for (int q = 1; q < NQ; ++q) { pre *= z[q]; o[q] = pre; }
    float suf = z[8];
    #pragma unroll
    for (int q = 7; q >= 1; --q) suf *= z[q];
    o[0] = suf;

    float* obase = out + (((size_t)b * NQ) * HOUT + (size_t)ho) * WOUT + wo;
    #pragma unroll
    for (int q = 0; q < NQ; ++q)
        obase[(size_t)q * HOUT * WOUT] = o[q];
}

extern "C" void kernel_launch(void* const* d_in, const int* in_sizes, int n_in,
                              void* d_out, int out_size, void* d_ws, size_t ws_size,
                              hipStream_t stream) {
    const float* x = (const float*)d_in[0];   // (4,1,384,384) f32
    const float* w = (const float*)d_in[1];   // (27,) f32
    float* out = (float*)d_out;               // (4,9,128,128) f32
    (void)in_sizes; (void)n_in; (void)out_size; (void)d_ws; (void)ws_size;
    quanconv_kernel<<<dim3(4 * HOUT), dim3(128), 0, stream>>>(x, w, out);
}